// SpecAugment_62526133895490
// MI455X (gfx1250) — compile-verified
//
#include <hip/hip_runtime.h>
#include <stdint.h>
#include <math.h>

// SpecAugment for MI455X (gfx1250).
// Memory-bound: 524 MB total traffic -> ~22.5 us floor at 23.3 TB/s; WMMA is
// inapplicable (no contraction). Optimization targets: B128 nontemporal
// loads/stores, wave-uniform skip of time-masked rows (saves ~20% reads),
// loop-invariant mask state per wave.
//
// Layout facts exploited by the apply kernel:
//  - row = b*4000 + t; each wave handles 4 consecutive rows starting at a
//    multiple of 4; since 4000 % 4 == 0 those rows share one sample b, so the
//    time-mask bounds and the b/t decomposition are loop-invariant per wave.
//  - F=128 floats/row = 32 lanes x float4 -> one row per wave, branch on the
//    time mask is wave-uniform (no divergence).

typedef float  float4v __attribute__((ext_vector_type(4)));
typedef int    int4v   __attribute__((ext_vector_type(4)));

#define B_DIM 128
#define T_DIM 4000
#define F_DIM 128
#define ROWS  (B_DIM * T_DIM)          // 512000
#define ROWS_PER_BLOCK 32              // 8 waves * 4 rows
#define GRID  (ROWS / ROWS_PER_BLOCK)  // 16000

// ---------------- Streaming apply (hot kernel, listed first) ----------------
__global__ __launch_bounds__(256) void specaug_apply(const float* __restrict__ x,
                                                     float* __restrict__ out,
                                                     const float* __restrict__ fmask_keep,
                                                     const int4v* __restrict__ tmask_bounds) {
  const int lane = threadIdx.x & 31;
  const int wave = threadIdx.x >> 5;
  const int row0 = blockIdx.x * ROWS_PER_BLOCK + wave * 4;

  // Loop-invariant per wave: sample index, base time, time-mask bounds.
  const unsigned ub = (unsigned)row0 / (unsigned)T_DIM;
  const int t0 = row0 - (int)(ub * (unsigned)T_DIM);
  const int4v tb = tmask_bounds[ub];

  // Per-lane frequency multiplier (512 B table, cache-hot -> regular load).
  const float4v fmul = *(const float4v*)(fmask_keep + lane * 4);

  const size_t base = (size_t)row0 * F_DIM + (size_t)(lane * 4);
  const float* xp = x + base;
  float*       op = out + base;

#pragma unroll
  for (int r = 0; r < 4; ++r) {
    const int t = t0 + r;
    const bool tmasked = (t >= tb.x && t < tb.y) || (t >= tb.z && t < tb.w);

    if (tmasked) {
      // Entire row zeroed: skip the read, stream zeros.
      float4v z = {0.0f, 0.0f, 0.0f, 0.0f};
      __builtin_nontemporal_store(z, (float4v*)op);
    } else {
      float4v v = __builtin_nontemporal_load((const float4v*)xp);
      v *= fmul;
      __builtin_nontemporal_store(v, (float4v*)op);
    }
    xp += F_DIM;
    op += F_DIM;
  }
}

// ---------------- Threefry-2x32 (JAX schedule: 20 rounds) ----------------
__device__ __forceinline__ void threefry2x32(uint32_t k0, uint32_t k1,
                                             uint32_t x0, uint32_t x1,
                                             uint32_t& o0, uint32_t& o1) {
  const uint32_t ks[3] = {k0, k1, k0 ^ k1 ^ 0x1BD11BDAu};
  x0 += ks[0];
  x1 += ks[1];
  const int rot_a[4] = {13, 15, 26, 6};
  const int rot_b[4] = {17, 29, 16, 24};
#pragma unroll
  for (int i = 0; i < 5; ++i) {
#pragma unroll
    for (int r = 0; r < 4; ++r) {
      const int d = (i & 1) ? rot_b[r] : rot_a[r];
      x0 += x1;
      x1 = (x1 << d) | (x1 >> (32 - d));
      x1 ^= x0;
    }
    x0 += ks[(i + 1) % 3];
    x1 += ks[(i + 2) % 3] + (uint32_t)(i + 1);
  }
  o0 = x0;
  o1 = x1;
}

__device__ __forceinline__ float bits_to_unit_float(uint32_t bits) {
  // JAX float32 uniform: bitcast((bits>>9) | 0x3f800000) - 1.0
  return __uint_as_float((bits >> 9) | 0x3f800000u) - 1.0f;
}

// flat random_bits element k of a length-256 draw (counts [0..255]):
// x0 = [0..127], x1 = [128..255]; out = concat(out0_vec, out1_vec)
__device__ __forceinline__ float flat_uniform256(uint32_t k0, uint32_t k1, int idx) {
  uint32_t o0, o1;
  if (idx < 128) {
    threefry2x32(k0, k1, (uint32_t)idx, (uint32_t)(idx + 128), o0, o1);
    return bits_to_unit_float(o0);
  } else {
    threefry2x32(k0, k1, (uint32_t)(idx - 128), (uint32_t)idx, o0, o1);
    return bits_to_unit_float(o1);
  }
}

// ---------------- Setup: masks into workspace (runs once, 1 block) ----------
// fmask_keep[f] in {0,1}; tmask_bounds[b] = (ts0, te0, ts1, te1)
__global__ __launch_bounds__(128) void specaug_setup(const int* __restrict__ x_len,
                                                     float* __restrict__ fmask_keep,
                                                     int4v* __restrict__ tmask_bounds) {
  const int tid = threadIdx.x;  // 0..127

  // split(key(42), 3): counts [0..5] -> lanes (i, i+3); keys reshape (3,2)
  uint32_t a0, b0, a1, b1, a2, b2;
  threefry2x32(0u, 42u, 0u, 3u, a0, b0);
  threefry2x32(0u, 42u, 1u, 4u, a1, b1);
  threefry2x32(0u, 42u, 2u, 5u, a2, b2);
  const uint32_t kf0 = a0, kf1 = a1;   // freq key
  const uint32_t kv0 = a2, kv1 = b0;   // time value key
  const uint32_t ks0 = b1, ks1 = b2;   // time start key

  // ---- frequency mask: uniform(kf, (2,2)); counts [0..3] -> lanes (i, i+2)
  uint32_t fa0, fb0, fa1, fb1;
  threefry2x32(kf0, kf1, 0u, 2u, fa0, fb0);
  threefry2x32(kf0, kf1, 1u, 3u, fa1, fb1);
  const float u00 = bits_to_unit_float(fa0);  // uf[0,0] -> f_val0
  const float u01 = bits_to_unit_float(fa1);  // uf[0,1] -> f_start0
  const float u10 = bits_to_unit_float(fb0);  // uf[1,0] -> f_val1
  const float u11 = bits_to_unit_float(fb1);  // uf[1,1] -> f_start1

  const float fval0 = u00 * 27.0f;
  const float fst0  = u01 * (128.0f - fval0);
  const float fval1 = u10 * 27.0f;
  const float fst1  = u11 * (128.0f - fval1);
  const float fs0 = floorf(fst0), fe0 = floorf(fst0 + fval0);
  const float fs1 = floorf(fst1), fe1 = floorf(fst1 + fval1);

  const float fi = (float)tid;
  const bool fmasked = (fi >= fs0 && fi < fe0) || (fi >= fs1 && fi < fe1);
  fmask_keep[tid] = fmasked ? 0.0f : 1.0f;

  // ---- time masks for sample b = tid (uniform(kv,(128,2)), uniform(ks,(128,2)))
  const int b = tid;
  const float xl = (float)x_len[b];
  const float t_param = floorf(0.25f * xl);

  const float uv0 = flat_uniform256(kv0, kv1, 2 * b);
  const float uv1 = flat_uniform256(kv0, kv1, 2 * b + 1);
  const float us0 = flat_uniform256(ks0, ks1, 2 * b);
  const float us1 = flat_uniform256(ks0, ks1, 2 * b + 1);

  const float tv0 = uv0 * t_param;
  const float tv1 = uv1 * t_param;
  const float tb0 = us0 * (xl - tv0);
  const float tb1 = us1 * (xl - tv1);

  int4v bounds;
  bounds.x = (int)floorf(tb0);
  bounds.y = (int)floorf(tb0 + tv0);
  bounds.z = (int)floorf(tb1);
  bounds.w = (int)floorf(tb1 + tv1);
  tmask_bounds[b] = bounds;
}

extern "C" void kernel_launch(void* const* d_in, const int* in_sizes, int n_in,
                              void* d_out, int out_size, void* d_ws, size_t ws_size,
                              hipStream_t stream) {
  const float* x     = (const float*)d_in[0];
  const int*   x_len = (const int*)d_in[1];
  float*       out   = (float*)d_out;

  float* fmask_keep   = (float*)d_ws;                                  // 128 floats
  int4v* tmask_bounds = (int4v*)((char*)d_ws + F_DIM * sizeof(float)); // 128 int4 (16B aligned)

  specaug_setup<<<1, 128, 0, stream>>>(x_len, fmask_keep, tmask_bounds);
  specaug_apply<<<GRID, 256, 0, stream>>>(x, out, fmask_keep, tmask_bounds);
}